// SparseProjector_86870008529137
// MI455X (gfx1250) — compile-verified
//
#include <hip/hip_runtime.h>
#include <stdint.h>

// Problem constants (match reference).
#define SRC_SIZE 262144
#define DST_SIZE 65536
#define N_EDGES  2097152
#define BATCH    4
#define FEAT     32

#define TILE     512   // edges per staged tile (2 KB per edge array)
#define BLOCK    256   // 8 waves of 32
#define CHUNKS   4     // tiles processed per block (double-buffered)

// ---------------------------------------------------------------------------
// Device helpers: guaranteed-native CDNA5 instructions via inline asm.
// ---------------------------------------------------------------------------

// No-return f32 atomic add, device scope (RMW resolves in L2 — out/norm are
// L2-resident). Avoids HIP's CAS-loop fallback.
__device__ __forceinline__ void atomic_add_f32_dev(float* addr, float val) {
  asm volatile("global_atomic_add_f32 %0, %1, off scope:SCOPE_DEV"
               :: "v"(addr), "v"(val)
               : "memory");
}

// Async global -> LDS copy, 16B per lane (GVS addressing: saddr(64b SGPR) +
// voffset(32b VGPR)). Tracked by ASYNCcnt.
__device__ __forceinline__ void async_load_b128_to_lds(uint32_t lds_addr,
                                                       uint32_t voffset,
                                                       const void* saddr) {
  asm volatile("global_load_async_to_lds_b128 %0, %1, %2"
               :: "v"(lds_addr), "v"(voffset), "s"(saddr)
               : "memory");
}

__device__ __forceinline__ void wait_asynccnt0() {
  asm volatile("s_wait_asynccnt 0" ::: "memory");
}

// ---------------------------------------------------------------------------
// Phase 1: norm[d] = sum_e { dst[e]==d } weights[e]   (256 KB, lives in L2)
// ---------------------------------------------------------------------------
__global__ __launch_bounds__(BLOCK) void norm_kernel(
    const float* __restrict__ weights,
    const int*   __restrict__ dst_idx,
    float*       __restrict__ norm) {
  int e = blockIdx.x * BLOCK + threadIdx.x;
  if (e < N_EDGES) {
    atomic_add_f32_dev(&norm[dst_idx[e]], weights[e]);
  }
}

// ---------------------------------------------------------------------------
// Phase 2: out[b,dst,:] += (w[e]/(norm[dst]+1e-8)) * x[b,src,:]
//   - edge streams staged into double-buffered LDS with b128 async loads;
//     next tile's staging overlaps current tile's scatter (ASYNCcnt drained
//     only after the scatter work)
//   - wave-per-edge; lane = batch(2b) x feature-group(3b); float4 gather
//   - four device-scope f32 atomics per lane into L2-resident out rows
// ---------------------------------------------------------------------------
__global__ __launch_bounds__(BLOCK) void scatter_kernel(
    const float* __restrict__ x,
    const float* __restrict__ weights,
    const int*   __restrict__ src_idx,
    const int*   __restrict__ dst_idx,
    const float* __restrict__ norm,
    float*       __restrict__ out) {
  __shared__ __align__(16) float s_w[2][TILE];
  __shared__ __align__(16) int   s_src[2][TILE];
  __shared__ __align__(16) int   s_dst[2][TILE];

  const int tid = threadIdx.x;
  const int blockEdgeBase = blockIdx.x * (TILE * CHUNKS);

  // LDS byte addresses of the two staging buffers.
  uint32_t lw[2], ls[2], ld[2];
  #pragma unroll
  for (int bfr = 0; bfr < 2; ++bfr) {
    lw[bfr] = (uint32_t)(uintptr_t)&s_w[bfr][0];
    ls[bfr] = (uint32_t)(uintptr_t)&s_src[bfr][0];
    ld[bfr] = (uint32_t)(uintptr_t)&s_dst[bfr][0];
  }

  // Issue async staging of one 512-edge tile into buffer `bfr`.
  // 128 b128 ops per 2KB array: tid 0..127 -> w and dst, tid 128..255 -> src.
  auto stage_tile = [&](int tileIdx, int bfr) {
    const int base = blockEdgeBase + tileIdx * TILE;
    if (tid < 128) {
      const uint32_t off = (uint32_t)tid * 16u;
      async_load_b128_to_lds(lw[bfr] + off, off, weights + base);
      async_load_b128_to_lds(ld[bfr] + off, off, dst_idx + base);
    } else {
      const uint32_t off = (uint32_t)(tid - 128) * 16u;
      async_load_b128_to_lds(ls[bfr] + off, off, src_idx + base);
    }
  };

  const int wave = tid >> 5;        // 0..7
  const int lane = tid & 31;
  const int b    = lane >> 3;       // 0..3  (batch)
  const int fg   = lane & 7;        // 0..7  (group of 4 features)

  // Prologue: stage tile 0.
  stage_tile(0, 0);
  wait_asynccnt0();
  __syncthreads();

  for (int t = 0; t < CHUNKS; ++t) {
    const int cur = t & 1;

    // Overlap: kick off next tile's staging into the other buffer. That
    // buffer was last read in iteration t-1, fenced by its trailing
    // wait+barrier, so the writes cannot race readers.
    if (t + 1 < CHUNKS) stage_tile(t + 1, cur ^ 1);

    // ---- Normalize weights in LDS (norm gather hits L2) ----
    #pragma unroll
    for (int i = 0; i < TILE / BLOCK; ++i) {
      const int e = tid + i * BLOCK;
      const float n = norm[s_dst[cur][e]];
      s_w[cur][e] = s_w[cur][e] / (n + 1e-8f);
    }
    __syncthreads();

    // ---- Wave-per-edge scatter ----
    for (int i = wave; i < TILE; i += BLOCK / 32) {
      const int   src = s_src[cur][i];   // LDS broadcast reads
      const int   dst = s_dst[cur][i];
      const float w   = s_w[cur][i];

      const float4 v = *(const float4*)(
          x + ((size_t)b * SRC_SIZE + (size_t)src) * FEAT + (size_t)fg * 4);
      float* o = out + ((size_t)b * DST_SIZE + (size_t)dst) * FEAT + (size_t)fg * 4;

      atomic_add_f32_dev(o + 0, w * v.x);
      atomic_add_f32_dev(o + 1, w * v.y);
      atomic_add_f32_dev(o + 2, w * v.z);
      atomic_add_f32_dev(o + 3, w * v.w);
    }
    __syncthreads();

    // Drain the (long-since overlapped) staging of the next tile, then make
    // its LDS contents visible to all waves.
    if (t + 1 < CHUNKS) {
      wait_asynccnt0();
      __syncthreads();
    }
  }
}

// ---------------------------------------------------------------------------
// Host entry
// ---------------------------------------------------------------------------
extern "C" void kernel_launch(void* const* d_in, const int* in_sizes, int n_in,
                              void* d_out, int out_size, void* d_ws, size_t ws_size,
                              hipStream_t stream) {
  const float* x       = (const float*)d_in[0];
  const float* weights = (const float*)d_in[1];
  const int*   src_idx = (const int*)d_in[2];
  const int*   dst_idx = (const int*)d_in[3];
  float* out  = (float*)d_out;
  float* norm = (float*)d_ws;   // DST_SIZE floats = 256 KB of scratch

  // Zero accumulators (graph-capture-safe async memsets).
  hipMemsetAsync(norm, 0, (size_t)DST_SIZE * sizeof(float), stream);
  hipMemsetAsync(out, 0, (size_t)out_size * sizeof(float), stream);

  norm_kernel<<<N_EDGES / BLOCK, BLOCK, 0, stream>>>(weights, dst_idx, norm);

  scatter_kernel<<<N_EDGES / (TILE * CHUNKS), BLOCK, 0, stream>>>(
      x, weights, src_idx, dst_idx, norm, out);
}